// Radix2ModGroup_70282844832377
// MI455X (gfx1250) — compile-verified
//
#include <hip/hip_runtime.h>
#include <hip/hip_bf16.h>
#include <stdint.h>

#define BLOCK 256

// 8x8 bit-matrix transpose of a u64 (bytes = rows). Hacker's Delight delta-swaps.
__device__ __forceinline__ uint64_t transpose8x8(uint64_t x) {
    uint64_t t;
    t = (x ^ (x >> 7))  & 0x00AA00AA00AA00AAULL; x ^= t ^ (t << 7);
    t = (x ^ (x >> 14)) & 0x0000CCCC0000CCCCULL; x ^= t ^ (t << 14);
    t = (x ^ (x >> 28)) & 0x00000000F0F0F0F0ULL; x ^= t ^ (t << 28);
    return x;
}

// CDNA5 async global->LDS copy, 16B per lane, tracked by ASYNCcnt.
__device__ __forceinline__ void async_copy_b128(const float* gsrc, float* ldst) {
    unsigned ldsoff = (unsigned)(uintptr_t)ldst;   // low 32 bits of generic addr = LDS byte offset
    asm volatile("global_load_async_to_lds_b128 %0, %1, off"
                 :: "v"(ldsoff), "v"(gsrc) : "memory");
}

__device__ __forceinline__ void wait_async0() {
    asm volatile("s_wait_asynccnt 0" ::: "memory");
}

__global__ __launch_bounds__(BLOCK)
void radix2_mod_group_kernel(const float* __restrict__ x,
                             float* __restrict__ out,
                             unsigned ngroups) {
    __shared__ __align__(16) float tile[BLOCK * 8];   // 8 KB/block (LDS is 320KB/WGP)
    const unsigned tid = threadIdx.x;
    const unsigned g   = blockIdx.x * BLOCK + tid;    // one group of 8 floats per lane
    if (g >= ngroups) return;

    const float* src  = x + (size_t)g * 8;
    float*       lbuf = &tile[tid * 8];

    // Stage this lane's 32 bytes through LDS via the async tensor-data path.
    async_copy_b128(src,     lbuf);
    async_copy_b128(src + 4, lbuf + 4);
    wait_async0();                                    // wave-scope: each lane reads only its own copy

    float4 a = *(const float4*)(lbuf);                // ds_load_b128 x2
    float4 b = *(const float4*)(lbuf + 4);
    float f[8] = {a.x, a.y, a.z, a.w, b.x, b.y, b.z, b.w};

    // Quantize: q = clip(round(x/0.05), -255, 255); pack |q| of value j into byte (7-j).
    uint64_t packed = 0;
    #pragma unroll
    for (int j = 0; j < 8; ++j) {
        float r = __builtin_rintf(f[j] * 20.0f);      // RNE, matches jnp.round
        float m = fminf(fabsf(r), 255.0f);
        packed |= (uint64_t)(unsigned)m << (8 * (7 - j));
    }

    // Priority layout: bit (8e + 7-j) = bit e of value j.
    // Higher bit position == higher keep-priority (exponent desc, index asc).
    uint64_t T = transpose8x8(packed);

    int c = __popcll(T);
    if (c > 12) {
        int n = c - 12;                               // drop the n lowest-priority set bits
        uint32_t w = (uint32_t)T;
        int pos = 0;
        int clo = __popc(w);
        if (n > clo) { n -= clo; pos = 32; w = (uint32_t)(T >> 32); }
        int cc;
        cc = __popc(w & 0xFFFFu); if (n > cc) { n -= cc; pos += 16; w >>= 16; }
        cc = __popc(w & 0x00FFu); if (n > cc) { n -= cc; pos += 8;  w >>= 8;  }
        cc = __popc(w & 0x000Fu); if (n > cc) { n -= cc; pos += 4;  w >>= 4;  }
        cc = __popc(w & 0x0003u); if (n > cc) { n -= cc; pos += 2;  w >>= 2;  }
        cc = (int)(w & 1u);       if (n > cc) { pos += 1; }
        T &= ~((2ULL << pos) - 1ULL);                 // clear bits 0..pos (exactly c-12 set bits)
    }

    // Back to value-major: byte (7-j) = kept magnitude of value j.
    uint64_t R = transpose8x8(T);

    float o[8];
    #pragma unroll
    for (int j = 0; j < 8; ++j) {
        unsigned m = (unsigned)(R >> (8 * (7 - j))) & 0xFFu;
        o[j] = __builtin_copysignf((float)m * 0.05f, f[j]);
    }

    float4* dst = (float4*)(out + (size_t)g * 8);     // global_store_b128 x2
    dst[0] = make_float4(o[0], o[1], o[2], o[3]);
    dst[1] = make_float4(o[4], o[5], o[6], o[7]);
}

extern "C" void kernel_launch(void* const* d_in, const int* in_sizes, int n_in,
                              void* d_out, int out_size, void* d_ws, size_t ws_size,
                              hipStream_t stream) {
    const float* x   = (const float*)d_in[0];
    float*       out = (float*)d_out;
    unsigned n       = (unsigned)in_sizes[0];
    unsigned ngroups = n / 8;                          // 4096*4096/8 = 2,097,152
    unsigned blocks  = (ngroups + BLOCK - 1) / BLOCK;  // 8192 blocks of 256 (8 waves ea.)
    radix2_mod_group_kernel<<<blocks, BLOCK, 0, stream>>>(x, out, ngroups);
}